// TransformerNet_10685878633053
// MI455X (gfx1250) — compile-verified
//
#include <hip/hip_runtime.h>

// ---------------- problem constants ----------------
#define BDIM 16
#define LDIM 50
#define KDIM 20
#define EDIM 64
#define HDIM 2
#define DH   32
#define CATV 2048
#define SDIM 2049          // CAT_V + 1
#define SP   2064          // padded to 129 * 16
#define NT   129           // 16-row tiles over SP
#define QT2  65            // ceil(NT/2): query-tile PAIRS per (b,h)

typedef float v2f __attribute__((ext_vector_type(2)));
typedef float v8f __attribute__((ext_vector_type(8)));

// D = A(16x4) * B(4x16) + C(16x16), all fp32, wave32 WMMA
__device__ __forceinline__ v8f wmma_f32(v2f a, v2f b, v8f c) {
  return __builtin_amdgcn_wmma_f32_16x16x4_f32(false, a, false, b, (short)0, c,
                                               false, false);
}

// reductions across the 16 lanes of a half-wave (rows of a C fragment live
// entirely within one half; xor masks < 16 never cross halves)
__device__ __forceinline__ float hmax16(float v) {
  v = fmaxf(v, __shfl_xor(v, 1, 32));
  v = fmaxf(v, __shfl_xor(v, 2, 32));
  v = fmaxf(v, __shfl_xor(v, 4, 32));
  v = fmaxf(v, __shfl_xor(v, 8, 32));
  return v;
}
__device__ __forceinline__ float hsum16(float v) {
  v += __shfl_xor(v, 1, 32);
  v += __shfl_xor(v, 2, 32);
  v += __shfl_xor(v, 4, 32);
  v += __shfl_xor(v, 8, 32);
  return v;
}

// ---------------- 1: build x[b,s,e] base (id / cat emb, zero padding) -------
__global__ void init_x_kernel(const int* __restrict__ id_arr,
                              const float* __restrict__ id_emb,
                              const float* __restrict__ cat_emb,
                              float* __restrict__ X) {
  int idx = blockIdx.x * blockDim.x + threadIdx.x;   // (b*SP + s)*E + e
  int e = idx % EDIM;
  int t = idx / EDIM;
  int s = t % SP;
  int b = t / SP;
  float v;
  if (s == 0)          v = id_emb[(size_t)id_arr[b] * EDIM + e];
  else if (s < SDIM)   v = cat_emb[(s - 1) * EDIM + e];
  else                 v = 0.0f;
  X[idx] = v;
}

// ---------------- 2: scatter dt+amt embeddings by category id --------------
// categories are distinct within each basket, so adding dt_e once per (l,k)
// equals the clipped-onehot "present" scatter of the reference.
__global__ void scatter_kernel(const int* __restrict__ cat_arr,
                               const int* __restrict__ dt_arr,
                               const int* __restrict__ amount_arr,
                               const float* __restrict__ amt_emb,
                               const float* __restrict__ dt_emb,
                               float* __restrict__ X) {
  int idx = blockIdx.x * blockDim.x + threadIdx.x;   // (((b*L)+l)*K+k)*E + e
  int e = idx % EDIM;
  int t = idx / EDIM;
  int k = t % KDIM; t /= KDIM;
  int lb = t % LDIM;
  int b = t / LDIM;
  int base = (b * LDIM + lb) * KDIM + k;
  int cat = cat_arr[base];
  int amt = amount_arr[base];
  int dt  = dt_arr[b * LDIM + lb];
  float v = amt_emb[amt * EDIM + e] + dt_emb[dt * EDIM + e];
  atomicAdd(&X[((size_t)b * SP + 1 + cat) * EDIM + e], v);
}

// ---------------- 3: qkv = x @ in_proj_w^T + b  (one wave per 16x16 tile) ---
__global__ void qkv_kernel(const float* __restrict__ X,
                           const float* __restrict__ W,       // [192,64]
                           const float* __restrict__ bias,    // [192]
                           float* __restrict__ Q, float* __restrict__ Kb,
                           float* __restrict__ Vb) {
  int id = blockIdx.x;
  int ct = id % 12; id /= 12;          // 192 / 16 col tiles
  int rt = id % NT;
  int b  = id / NT;
  int lane = threadIdx.x & 31, half = lane >> 4, l = lane & 15;
  int rb = b * SP + rt * 16;
  int n0 = ct * 16;
  v8f c = {0.f,0.f,0.f,0.f,0.f,0.f,0.f,0.f};
#pragma unroll
  for (int kk = 0; kk < 16; ++kk) {
    int k = kk * 4 + half * 2;
    v2f a, bb;
    a.x  = X[(rb + l) * EDIM + k];
    a.y  = X[(rb + l) * EDIM + k + 1];
    bb.x = W[(n0 + l) * EDIM + k];
    bb.y = W[(n0 + l) * EDIM + k + 1];
    c = wmma_f32(a, bb, c);
  }
  int n = n0 + l;
  float bv = bias[n];
  float* dst; int nn;
  if (n < 64)       { dst = Q;  nn = n; }
  else if (n < 128) { dst = Kb; nn = n - 64; }
  else              { dst = Vb; nn = n - 128; }
#pragma unroll
  for (int r = 0; r < 8; ++r)
    dst[(rb + r + 8 * half) * EDIM + nn] = c[r] + bv;
}

// ---------------- 4: flash attention, one wave per (b, h, PAIR of 16-row
// query tiles).  K/V tile fragments are loaded once and shared by both query
// tiles, halving vmem traffic per WMMA.  Next key tile is prefetched
// (global_prefetch_b8) while the current one computes. -----------------------
__global__ void attn_kernel(const float* __restrict__ Q,
                            const float* __restrict__ Kb,
                            const float* __restrict__ Vb,
                            float* __restrict__ O) {
  __shared__ float pl0[256];           // P staging tile 0 (C-layout -> A-layout)
  __shared__ float pl1[256];           // P staging tile 1
  int id = blockIdx.x;
  int qp = id % QT2; id /= QT2;
  int h  = id % HDIM;
  int b  = id / HDIM;
  int lane = threadIdx.x & 31, half = lane >> 4, l = lane & 15;
  int hd0 = h * DH;
  int qt0 = qp * 2;
  int qt1 = (qt0 + 1 < NT) ? qt0 + 1 : qt0;   // odd NT: last pair duplicates
  int qrb0 = b * SP + qt0 * 16;
  int qrb1 = b * SP + qt1 * 16;

  float qa0[16], qa1[16];
#pragma unroll
  for (int kk = 0; kk < 8; ++kk) {
    int k = kk * 4 + half * 2;
    qa0[2 * kk]     = Q[(qrb0 + l) * EDIM + hd0 + k];
    qa0[2 * kk + 1] = Q[(qrb0 + l) * EDIM + hd0 + k + 1];
    qa1[2 * kk]     = Q[(qrb1 + l) * EDIM + hd0 + k];
    qa1[2 * kk + 1] = Q[(qrb1 + l) * EDIM + hd0 + k + 1];
  }
  float m0[8], l0[8], m1[8], l1[8];
  v8f o00 = {0.f,0.f,0.f,0.f,0.f,0.f,0.f,0.f};
  v8f o01 = {0.f,0.f,0.f,0.f,0.f,0.f,0.f,0.f};
  v8f o10 = {0.f,0.f,0.f,0.f,0.f,0.f,0.f,0.f};
  v8f o11 = {0.f,0.f,0.f,0.f,0.f,0.f,0.f,0.f};
#pragma unroll
  for (int r = 0; r < 8; ++r) {
    m0[r] = -1e30f; l0[r] = 0.0f;
    m1[r] = -1e30f; l1[r] = 0.0f;
  }
  const float IS = 0.17677669529663687f;   // 1/sqrt(32)

  for (int kt = 0; kt < NT; ++kt) {
    int krb = b * SP + kt * 16;
    // prefetch next key tile rows (fire-and-forget, no LOADcnt)
    if (kt + 1 < NT) {
      int nrb = krb + 16;
      __builtin_prefetch(&Kb[(nrb + l) * EDIM + hd0], 0, 0);
      __builtin_prefetch(&Vb[(nrb + l) * EDIM + hd0], 0, 0);
    }
    // K^T fragments for this tile, shared by both query tiles
    float kf[16];
#pragma unroll
    for (int kk = 0; kk < 8; ++kk) {
      int k = kk * 4 + half * 2;
      kf[2 * kk]     = Kb[(krb + l) * EDIM + hd0 + k];
      kf[2 * kk + 1] = Kb[(krb + l) * EDIM + hd0 + k + 1];
    }
    v8f sc0 = {0.f,0.f,0.f,0.f,0.f,0.f,0.f,0.f};
    v8f sc1 = {0.f,0.f,0.f,0.f,0.f,0.f,0.f,0.f};
#pragma unroll
    for (int kk = 0; kk < 8; ++kk) {        // scores = Q(16x32) @ K^T(32x16)
      v2f a0, a1, bb;
      a0.x = qa0[2 * kk]; a0.y = qa0[2 * kk + 1];
      a1.x = qa1[2 * kk]; a1.y = qa1[2 * kk + 1];
      bb.x = kf[2 * kk];  bb.y = kf[2 * kk + 1];
      sc0 = wmma_f32(a0, bb, sc0);
      sc1 = wmma_f32(a1, bb, sc1);
    }
    bool valid = (kt * 16 + l) < SDIM;       // mask padded key columns
#pragma unroll
    for (int r = 0; r < 8; ++r) {
      // --- query tile 0 online softmax ---
      float s  = valid ? sc0[r] * IS : -1e30f;
      float nm = fmaxf(m0[r], hmax16(s));
      float cs = __expf(m0[r] - nm);
      float p  = __expf(s - nm);
      l0[r] = l0[r] * cs + hsum16(p);
      m0[r] = nm;
      o00[r] *= cs;  o01[r] *= cs;
      pl0[(r + 8 * half) * 16 + l] = p;
      // --- query tile 1 online softmax ---
      float s1  = valid ? sc1[r] * IS : -1e30f;
      float nm1 = fmaxf(m1[r], hmax16(s1));
      float cs1 = __expf(m1[r] - nm1);
      float p1  = __expf(s1 - nm1);
      l1[r] = l1[r] * cs1 + hsum16(p1);
      m1[r] = nm1;
      o10[r] *= cs1;  o11[r] *= cs1;
      pl1[(r + 8 * half) * 16 + l] = p1;
    }
    __syncthreads();
#pragma unroll
    for (int kk = 0; kk < 4; ++kk) {         // O += P(16x16) @ V(16x32)
      int kr = kk * 4 + half * 2;
      v2f a0, a1, b0, b1;
      a0.x = pl0[l * 16 + kr];
      a0.y = pl0[l * 16 + kr + 1];
      a1.x = pl1[l * 16 + kr];
      a1.y = pl1[l * 16 + kr + 1];
      b0.x = Vb[(krb + kr) * EDIM + hd0 + l];
      b0.y = Vb[(krb + kr + 1) * EDIM + hd0 + l];
      b1.x = Vb[(krb + kr) * EDIM + hd0 + 16 + l];
      b1.y = Vb[(krb + kr + 1) * EDIM + hd0 + 16 + l];
      o00 = wmma_f32(a0, b0, o00);
      o01 = wmma_f32(a0, b1, o01);
      o10 = wmma_f32(a1, b0, o10);
      o11 = wmma_f32(a1, b1, o11);
    }
    __syncthreads();
  }
#pragma unroll
  for (int r = 0; r < 8; ++r) {
    float inv0 = 1.0f / l0[r];
    float inv1 = 1.0f / l1[r];
    O[(qrb0 + r + 8 * half) * EDIM + hd0 + l]      = o00[r] * inv0;
    O[(qrb0 + r + 8 * half) * EDIM + hd0 + 16 + l] = o01[r] * inv0;
    O[(qrb1 + r + 8 * half) * EDIM + hd0 + l]      = o10[r] * inv1;
    O[(qrb1 + r + 8 * half) * EDIM + hd0 + 16 + l] = o11[r] * inv1;
  }
}

// ---------------- 5: out_proj + residual + LayerNorm1 ----------------------
__global__ void oproj_ln1_kernel(const float* __restrict__ O,
                                 const float* __restrict__ X,
                                 const float* __restrict__ W,    // [64,64]
                                 const float* __restrict__ bias,
                                 const float* __restrict__ lw,
                                 const float* __restrict__ lb,
                                 float* __restrict__ X2) {
  int id = blockIdx.x;
  int rt = id % NT;
  int b  = id / NT;
  int lane = threadIdx.x & 31, half = lane >> 4, l = lane & 15;
  int rb = b * SP + rt * 16;

  float oa[32];
#pragma unroll
  for (int kk = 0; kk < 16; ++kk) {
    int k = kk * 4 + half * 2;
    oa[2 * kk]     = O[(rb + l) * EDIM + k];
    oa[2 * kk + 1] = O[(rb + l) * EDIM + k + 1];
  }
  float val[4][8];
#pragma unroll
  for (int t = 0; t < 4; ++t) {
    v8f c = {0.f,0.f,0.f,0.f,0.f,0.f,0.f,0.f};
#pragma unroll
    for (int kk = 0; kk < 16; ++kk) {
      int k = kk * 4 + half * 2;
      v2f a, bb;
      a.x = oa[2 * kk]; a.y = oa[2 * kk + 1];
      bb.x = W[(t * 16 + l) * EDIM + k];
      bb.y = W[(t * 16 + l) * EDIM + k + 1];
      c = wmma_f32(a, bb, c);
    }
    float bv = bias[t * 16 + l];
#pragma unroll
    for (int r = 0; r < 8; ++r)
      val[t][r] = c[r] + bv + X[(rb + r + 8 * half) * EDIM + t * 16 + l];
  }
#pragma unroll
  for (int r = 0; r < 8; ++r) {
    float ps = val[0][r] + val[1][r] + val[2][r] + val[3][r];
    float p2 = val[0][r]*val[0][r] + val[1][r]*val[1][r] +
               val[2][r]*val[2][r] + val[3][r]*val[3][r];
    float mu  = hsum16(ps) * (1.0f / 64.0f);
    float var = hsum16(p2) * (1.0f / 64.0f) - mu * mu;
    float rstd = rsqrtf(var + 1e-5f);
#pragma unroll
    for (int t = 0; t < 4; ++t) {
      int col = t * 16 + l;
      X2[(rb + r + 8 * half) * EDIM + col] =
          (val[t][r] - mu) * rstd * lw[col] + lb[col];
    }
  }
}

// ---------------- 6: FFN + residual + LayerNorm2 + mean_E -> h -------------
__global__ void ffn_ln2_kernel(const float* __restrict__ X2,
                               const float* __restrict__ W1,
                               const float* __restrict__ b1,
                               const float* __restrict__ W2,
                               const float* __restrict__ b2,
                               const float* __restrict__ lw,
                               const float* __restrict__ lb,
                               float* __restrict__ Hout) {
  __shared__ float g[16 * 64];        // relu(x@ff1^T+b1) staging
  int id = blockIdx.x;
  int rt = id % NT;
  int b  = id / NT;
  int lane = threadIdx.x & 31, half = lane >> 4, l = lane & 15;
  int rb = b * SP + rt * 16;

  float xa[32];
#pragma unroll
  for (int kk = 0; kk < 16; ++kk) {
    int k = kk * 4 + half * 2;
    xa[2 * kk]     = X2[(rb + l) * EDIM + k];
    xa[2 * kk + 1] = X2[(rb + l) * EDIM + k + 1];
  }
#pragma unroll
  for (int t = 0; t < 4; ++t) {
    v8f c = {0.f,0.f,0.f,0.f,0.f,0.f,0.f,0.f};
#pragma unroll
    for (int kk = 0; kk < 16; ++kk) {
      int k = kk * 4 + half * 2;
      v2f a, bb;
      a.x = xa[2 * kk]; a.y = xa[2 * kk + 1];
      bb.x = W1[(t * 16 + l) * EDIM + k];
      bb.y = W1[(t * 16 + l) * EDIM + k + 1];
      c = wmma_f32(a, bb, c);
    }
    float bv = b1[t * 16 + l];
#pragma unroll
    for (int r = 0; r < 8; ++r)
      g[(r + 8 * half) * 64 + t * 16 + l] = fmaxf(c[r] + bv, 0.0f);
  }
  __syncthreads();

  float val[4][8];
#pragma unroll
  for (int t = 0; t < 4; ++t) {
    v8f c = {0.f,0.f,0.f,0.f,0.f,0.f,0.f,0.f};
#pragma unroll
    for (int kk = 0; kk < 16; ++kk) {
      int k = kk * 4 + half * 2;
      v2f a, bb;
      a.x = g[l * 64 + k];
      a.y = g[l * 64 + k + 1];
      bb.x = W2[(t * 16 + l) * EDIM + k];
      bb.y = W2[(t * 16 + l) * EDIM + k + 1];
      c = wmma_f32(a, bb, c);
    }
    float bv = b2[t * 16 + l];
#pragma unroll
    for (int r = 0; r < 8; ++r)
      val[t][r] = c[r] + bv + X2[(rb + r + 8 * half) * EDIM + t * 16 + l];
  }
#pragma unroll
  for (int r = 0; r < 8; ++r) {
    float ps = val[0][r] + val[1][r] + val[2][r] + val[3][r];
    float p2 = val[0][r]*val[0][r] + val[1][r]*val[1][r] +
               val[2][r]*val[2][r] + val[3][r]*val[3][r];
    float mu  = hsum16(ps) * (1.0f / 64.0f);
    float var = hsum16(p2) * (1.0f / 64.0f) - mu * mu;
    float rstd = rsqrtf(var + 1e-5f);
    float pf = 0.0f;
#pragma unroll
    for (int t = 0; t < 4; ++t) {
      int col = t * 16 + l;
      pf += (val[t][r] - mu) * rstd * lw[col] + lb[col];
    }
    float rmean = hsum16(pf) * (1.0f / 64.0f);
    int s = rt * 16 + r + 8 * half;
    if (l == 0 && s >= 1 && s < SDIM)
      Hout[b * CATV + (s - 1)] = rmean;
  }
}

// ---------------- 7/8: head GEMMs  out = act(A[16,2048] @ W^T + b) ----------
__global__ void head_kernel(const float* __restrict__ A,
                            const float* __restrict__ W,      // [2048,2048]
                            const float* __restrict__ bias,
                            float* __restrict__ out, int do_relu) {
  int n0 = blockIdx.x * 16;
  int lane = threadIdx.x & 31, half = lane >> 4, l = lane & 15;
  v8f c = {0.f,0.f,0.f,0.f,0.f,0.f,0.f,0.f};
#pragma unroll 8
  for (int kk = 0; kk < CATV / 4; ++kk) {
    int k = kk * 4 + half * 2;
    if ((kk & 31) == 0 && kk + 32 < CATV / 4) {
      // prefetch the streaming weight operand one 128-element chunk ahead
      __builtin_prefetch(&W[(n0 + l) * CATV + k + 128], 0, 0);
    }
    v2f a, bb;
    a.x  = A[l * CATV + k];
    a.y  = A[l * CATV + k + 1];
    bb.x = W[(n0 + l) * CATV + k];
    bb.y = W[(n0 + l) * CATV + k + 1];
    c = wmma_f32(a, bb, c);
  }
  float bv = bias[n0 + l];
#pragma unroll
  for (int r = 0; r < 8; ++r) {
    float v = c[r] + bv;
    if (do_relu) v = fmaxf(v, 0.0f);
    out[(r + 8 * half) * CATV + n0 + l] = v;
  }
}

// ---------------- launcher --------------------------------------------------
extern "C" void kernel_launch(void* const* d_in, const int* in_sizes, int n_in,
                              void* d_out, int out_size, void* d_ws,
                              size_t ws_size, hipStream_t stream) {
  const int*   cat_arr    = (const int*)d_in[0];
  const int*   dt_arr     = (const int*)d_in[1];
  const int*   amount_arr = (const int*)d_in[2];
  const int*   id_arr     = (const int*)d_in[3];
  const float* id_emb     = (const float*)d_in[4];
  const float* cat_emb    = (const float*)d_in[5];
  const float* amt_emb    = (const float*)d_in[6];
  const float* dt_emb     = (const float*)d_in[7];
  const float* in_proj_w  = (const float*)d_in[8];
  const float* in_proj_b  = (const float*)d_in[9];
  const float* out_proj_w = (const float*)d_in[10];
  const float* out_proj_b = (const float*)d_in[11];
  const float* ln1_w      = (const float*)d_in[12];
  const float* ln1_b      = (const float*)d_in[13];
  const float* ff1_w      = (const float*)d_in[14];
  const float* ff1_b      = (const float*)d_in[15];
  const float* ff2_w      = (const float*)d_in[16];
  const float* ff2_b      = (const float*)d_in[17];
  const float* ln2_w      = (const float*)d_in[18];
  const float* ln2_b      = (const float*)d_in[19];
  const float* lin1_w     = (const float*)d_in[20];
  const float* lin1_b     = (const float*)d_in[21];
  const float* lin2_w     = (const float*)d_in[22];
  const float* lin2_b     = (const float*)d_in[23];

  float* ws = (float*)d_ws;
  const size_t NX = (size_t)BDIM * SP * EDIM;   // 2,113,536 floats
  float* X  = ws;
  float* Q  = X  + NX;
  float* Kb = Q  + NX;
  float* Vb = Kb + NX;
  float* O  = Vb + NX;
  float* X2 = O  + NX;
  float* Hb = X2 + NX;                          // [16, 2048]
  float* H1 = Hb + (size_t)BDIM * CATV;         // [16, 2048]

  init_x_kernel<<<(BDIM * SP * EDIM) / 256, 256, 0, stream>>>(id_arr, id_emb,
                                                              cat_emb, X);
  scatter_kernel<<<(BDIM * LDIM * KDIM * EDIM) / 256, 256, 0, stream>>>(
      cat_arr, dt_arr, amount_arr, amt_emb, dt_emb, X);
  qkv_kernel<<<BDIM * NT * 12, 32, 0, stream>>>(X, in_proj_w, in_proj_b, Q, Kb,
                                                Vb);
  attn_kernel<<<BDIM * HDIM * QT2, 32, 0, stream>>>(Q, Kb, Vb, O);
  oproj_ln1_kernel<<<BDIM * NT, 32, 0, stream>>>(O, X, out_proj_w, out_proj_b,
                                                 ln1_w, ln1_b, X2);
  ffn_ln2_kernel<<<BDIM * NT, 32, 0, stream>>>(X2, ff1_w, ff1_b, ff2_w, ff2_b,
                                               ln2_w, ln2_b, Hb);
  head_kernel<<<CATV / 16, 32, 0, stream>>>(Hb, lin1_w, lin1_b, H1, 1);
  head_kernel<<<CATV / 16, 32, 0, stream>>>(H1, lin2_w, lin2_b, (float*)d_out,
                                            0);
}